// NonFirstLayerAggregator_53429393162861
// MI455X (gfx1250) — compile-verified
//
#include <hip/hip_runtime.h>

typedef __attribute__((ext_vector_type(2))) float v2f;
typedef __attribute__((ext_vector_type(8))) float v8f;

#define N_NODES 100000
#define NB      16384
#define S       32
#define D       64
#define SLOPE   0.2f

// ---------------------------------------------------------------------------
// Stage 1 (fused): H[c] = feat_table[t] @ W(head),  c = head*2 + t
//                  sdot[c][node] = H[c][node] . a_k(head)     (a_k = a[64:128])
// One wave per 16-row x 64-col output row-block: A fragments loaded once,
// 4 B fragments per k-step, 4 accumulators, 64 x V_WMMA_F32_16X16X4_F32.
// sdot fused in the epilogue via half-wave reductions over the C layout.
// 4 combos * 6250 row-blocks = 25000 waves (3125 blocks of 8 waves).
// ---------------------------------------------------------------------------
__global__ __launch_bounds__(256) void spgat_gemm_kernel(
    const float* __restrict__ feat, const float* __restrict__ W_bal,
    const float* __restrict__ W_unbal, const float* __restrict__ a_bal,
    const float* __restrict__ a_unbal, float* __restrict__ H,
    float* __restrict__ sdot)
{
  const int tid  = threadIdx.x;
  const int lane = tid & 31;
  const int wave = blockIdx.x * (blockDim.x >> 5) + (tid >> 5);

  const int c  = wave / 6250;            // combo 0..3 (head = c>>1, table = c&1)
  const int mt = wave % 6250;            // row-block 0..6249
  const int t  = c & 1;

  const float* A   = feat + (size_t)t * N_NODES * D;
  const float* Wm  = (c >> 1) ? W_unbal : W_bal;
  const float* ak  = (((c >> 1) ? a_unbal : a_bal)) + D;   // a_k = a[64:128]
  float*       out = H + (size_t)c * N_NODES * D;
  float*       sd  = sdot + (size_t)c * N_NODES;

  const int half  = lane >> 4;           // 0: K pair {k,k+1}, 1: K pair {k+2,k+3}
  const int m16   = lane & 15;
  const int mbase = mt * 16;

  v8f acc0 = {}, acc1 = {}, acc2 = {}, acc3 = {};
#pragma unroll
  for (int k = 0; k < D; k += 4) {
    const int kk = k + half * 2;
    // A fragment (shared by all 4 N-tiles): lanes 0-15 rows M=0..15 @ K={k,k+1},
    // lanes 16-31 same rows @ K={k+2,k+3}
    const float* arow = A + (size_t)(mbase + m16) * D + kk;
    v2f a;
    a.x = arow[0];
    a.y = arow[1];
    // B fragments for N-tiles 0..3: VGPR0 rows K={k,k+2}, VGPR1 rows K={k+1,k+3}
    const float* b0 = Wm + (size_t)kk * D + m16;
    const float* b1 = b0 + D;
    v2f bb0, bb1, bb2, bb3;
    bb0.x = b0[0];  bb0.y = b1[0];
    bb1.x = b0[16]; bb1.y = b1[16];
    bb2.x = b0[32]; bb2.y = b1[32];
    bb3.x = b0[48]; bb3.y = b1[48];
    acc0 = __builtin_amdgcn_wmma_f32_16x16x4_f32(false, a, false, bb0, (short)0, acc0, false, false);
    acc1 = __builtin_amdgcn_wmma_f32_16x16x4_f32(false, a, false, bb1, (short)0, acc1, false, false);
    acc2 = __builtin_amdgcn_wmma_f32_16x16x4_f32(false, a, false, bb2, (short)0, acc2, false, false);
    acc3 = __builtin_amdgcn_wmma_f32_16x16x4_f32(false, a, false, bb3, (short)0, acc3, false, false);
  }

  // Store C: VGPR r of tile nt holds (M = mbase+r+half*8, N = nt*16+m16)
#pragma unroll
  for (int r = 0; r < 8; ++r) {
    const size_t row = (size_t)(mbase + r + half * 8) * D;
    out[row + m16]      = acc0[r];
    out[row + 16 + m16] = acc1[r];
    out[row + 32 + m16] = acc2[r];
    out[row + 48 + m16] = acc3[r];
  }

  // Fused sdot: per row, dot with a_k. Each lane contributes its 4 columns,
  // then a 16-lane half-wave reduction (xor 8/4/2/1 stays within each half).
  const float ak0 = ak[m16], ak1 = ak[16 + m16], ak2 = ak[32 + m16], ak3 = ak[48 + m16];
#pragma unroll
  for (int r = 0; r < 8; ++r) {
    float p = acc0[r] * ak0 + acc1[r] * ak1 + acc2[r] * ak2 + acc3[r] * ak3;
#pragma unroll
    for (int o = 8; o; o >>= 1) p += __shfl_xor(p, o, 32);
    if (m16 == 0) sd[mbase + r + half * 8] = p;   // lane 0 -> M=r, lane 16 -> M=r+8
  }
}

// ---------------------------------------------------------------------------
// Stage 2: per (head, target) wave: gather 65 edge scores, softmax, aggregate.
// Edge order: j=0 self, j=1..32 neigh_pos, j=33..64 neigh_neg.
// Lane l holds edges j=l (A), j=32+l (B); lane 31 additionally j=64 (C).
// Output lanes own 2 channels each (float2).
// 32768 waves -> 4096 blocks of 8 waves.
// ---------------------------------------------------------------------------
__global__ __launch_bounds__(256) void spgat_attn_kernel(
    const int* __restrict__ nodes, const int* __restrict__ npos,
    const int* __restrict__ nneg, const float* __restrict__ H,
    const float* __restrict__ sdot, const float* __restrict__ a_bal,
    const float* __restrict__ a_unbal, float* __restrict__ out)
{
  const int tid  = threadIdx.x;
  const int lane = tid & 31;
  const int wave = blockIdx.x * (blockDim.x >> 5) + (tid >> 5);

  const int w = wave >> 14;      // 0 = bal head, 1 = unbal head
  const int b = wave & (NB - 1);

  // combo ids: bal head  -> H1 = table0@Wb (c=0), H2 = table1@Wb (c=1)
  //            unbal head-> H1 = table1@Wu (c=3), H2 = table0@Wu (c=2)
  const int c1 = 3 * w;
  const int c2 = w + 1;
  const float* H1 = H + (size_t)c1 * N_NODES * D;
  const float* H2 = H + (size_t)c2 * N_NODES * D;
  const float* s1 = sdot + (size_t)c1 * N_NODES;
  const float* s2 = sdot + (size_t)c2 * N_NODES;
  const float* aq = w ? a_unbal : a_bal;   // a_q = a[0:64]

  const int node = nodes[b];

  // qa = dot(H1[node], a_q), wave-wide
  const float2 q   = ((const float2*)(H1 + (size_t)node * D))[lane];
  const float2 aqv = ((const float2*)aq)[lane];
  float qa = q.x * aqv.x + q.y * aqv.y;
#pragma unroll
  for (int o = 16; o; o >>= 1) qa += __shfl_xor(qa, o, 32);

  const int* pos = npos + (size_t)b * S;
  const int* neg = nneg + (size_t)b * S;
  const int idxA = (lane == 0) ? node    : pos[lane - 1];   // j = lane
  const int idxB = (lane == 0) ? pos[31] : neg[lane - 1];   // j = 32 + lane
  const int idxC = neg[31];                                 // j = 64 (lane 31)

  const float sA = s1[idxA];
  const float sB = (lane == 0) ? s1[idxB] : s2[idxB];
  const float sC = s2[idxC];

  const float xA = qa + sA, xB = qa + sB, xC = qa + sC;
  const float eA = xA > 0.f ? xA : SLOPE * xA;
  const float eB = xB > 0.f ? xB : SLOPE * xB;
  const float eC = xC > 0.f ? xC : SLOPE * xC;

  float m = fmaxf(eA, eB);
  if (lane == 31) m = fmaxf(m, eC);
#pragma unroll
  for (int o = 16; o; o >>= 1) m = fmaxf(m, __shfl_xor(m, o, 32));

  const float pA = __expf(eA - m);
  const float pB = __expf(eB - m);
  const float pC = (lane == 31) ? __expf(eC - m) : 0.f;
  float sum = pA + pB + pC;
#pragma unroll
  for (int o = 16; o; o >>= 1) sum += __shfl_xor(sum, o, 32);
  const float inv = 1.f / sum;
  const float alA = pA * inv, alB = pB * inv, alC = pC * inv;

  float2 acc = make_float2(0.f, 0.f);
  for (int j = 0; j < 65; ++j) {
    float al;
    int idx;
    if (j < 32)       { al = __shfl(alA, j, 32);      idx = __shfl(idxA, j, 32); }
    else if (j < 64)  { al = __shfl(alB, j - 32, 32); idx = __shfl(idxB, j - 32, 32); }
    else              { al = __shfl(alC, 31, 32);     idx = __shfl(idxC, 31, 32); }
    const float* Hsrc = (j <= 32) ? H1 : H2;   // j==32 is pos[31] (edge-set 1)
    const float2 v = ((const float2*)(Hsrc + (size_t)idx * D))[lane];
    acc.x += al * v.x;
    acc.y += al * v.y;
  }

  ((float2*)(out + ((size_t)w * NB + b) * D))[lane] = acc;
}

extern "C" void kernel_launch(void* const* d_in, const int* in_sizes, int n_in,
                              void* d_out, int out_size, void* d_ws, size_t ws_size,
                              hipStream_t stream) {
  const int*   nodes   = (const int*)d_in[0];
  const int*   npos    = (const int*)d_in[1];
  const int*   nneg    = (const int*)d_in[2];
  const float* feat    = (const float*)d_in[3];
  const float* W_bal   = (const float*)d_in[4];
  const float* a_bal   = (const float*)d_in[5];
  const float* W_unbal = (const float*)d_in[6];
  const float* a_unbal = (const float*)d_in[7];
  float*       out     = (float*)d_out;

  float* H    = (float*)d_ws;                      // 4 * 100000 * 64 floats (102.4 MB)
  float* sdot = H + (size_t)4 * N_NODES * D;       // 4 * 100000 floats

  // 25000 row-block waves / 8 waves per block (exact)
  spgat_gemm_kernel<<<3125, 256, 0, stream>>>(feat, W_bal, W_unbal,
                                              a_bal, a_unbal, H, sdot);
  // 32768 (head,target) waves / 8
  spgat_attn_kernel<<<4096, 256, 0, stream>>>(nodes, npos, nneg, H, sdot,
                                              a_bal, a_unbal, out);
}